// MoBAGating_79087527788598
// MI455X (gfx1250) — compile-verified
//
#include <hip/hip_runtime.h>
#include <hip/hip_bf16.h>

typedef float v2f __attribute__((ext_vector_type(2)));
typedef float v8f __attribute__((ext_vector_type(8)));

#define BLK   512
#define TOPK  4
#define B_    4
#define H_    16
#define S_    8192
#define D_    128
#define NB    16      // S_/BLK

// ---------------------------------------------------------------------------
// Kernel 1: k_mean[bh, n, d] = mean over 512 rows of k.  Coalesced: 128
// threads own the 128 channels of a row; loop walks 512 rows.
// ---------------------------------------------------------------------------
__global__ __launch_bounds__(128) void kmean_kernel(const float* __restrict__ k,
                                                    float* __restrict__ kmean) {
    int bhn = blockIdx.x;            // 0 .. B*H*NB-1
    int tid = threadIdx.x;           // channel d
    int n   = bhn & (NB - 1);
    int bh  = bhn >> 4;
    const float* kp = k + ((size_t)bh * S_ + (size_t)n * BLK) * D_ + tid;
    float sum = 0.f;
#pragma unroll 8
    for (int s = 0; s < BLK; ++s) sum += kp[(size_t)s * D_];
    kmean[(size_t)bhn * D_ + tid] = sum * (1.0f / (float)BLK);
}

// ---------------------------------------------------------------------------
// Kernel 2: per (b,h, query-block qb): scores via V_WMMA_F32_16X16X4_F32,
// causal mask, per-row top-4.  4 waves / workgroup, 8 M-tiles per wave.
// ---------------------------------------------------------------------------
__global__ __launch_bounds__(128) void score_topk_kernel(const float* __restrict__ q,
                                                         const float* __restrict__ kmean,
                                                         int* __restrict__ topk_out) {
    __shared__ float lds_km[NB * D_];        // 8 KB: k_mean tile for this (b,h)
    __shared__ float lds_sc[4][16 * 17];     // per-wave 16x16 score tile, padded

    int qb   = blockIdx.x & (NB - 1);        // query block index (causal bound)
    int bh   = blockIdx.x >> 4;
    int tid  = threadIdx.x;
    int wave = tid >> 5;
    int lane = tid & 31;

    // Stage k_mean[bh] into LDS (coalesced)
    const float* km = kmean + (size_t)bh * NB * D_;
#pragma unroll
    for (int i = 0; i < 16; ++i) lds_km[tid + i * 128] = km[tid + i * 128];
    __syncthreads();

    // 32-bit WMMA A/B fragment addressing: lane L -> row (L&15),
    // float2 at col 4*c + 2*(L>>4)
    int mrow  = lane & 15;
    int khalf = (lane >> 4) << 1;

    // B = k_mean^T fragments, built once, reused for all 8 tiles
    v2f bfrag[32];
#pragma unroll
    for (int c = 0; c < 32; ++c)
        bfrag[c] = *(const v2f*)&lds_km[mrow * D_ + 4 * c + khalf];

    const float* qbh = q + (size_t)bh * S_ * D_;
    float* sc = &lds_sc[wave][0];
    int mhalf = (lane >> 4) << 3;            // C/D: VGPR r holds row r + 8*(lane/16)

    for (int t = 0; t < 8; ++t) {
        int rowBase = qb * BLK + (wave * 8 + t) * 16;
        const float* qrow = qbh + (size_t)(rowBase + mrow) * D_;

        v8f acc = {0.f, 0.f, 0.f, 0.f, 0.f, 0.f, 0.f, 0.f};
#pragma unroll
        for (int c = 0; c < 32; ++c) {
            v2f a = *(const v2f*)(qrow + 4 * c + khalf);
            acc = __builtin_amdgcn_wmma_f32_16x16x4_f32(
                      false, a, false, bfrag[c], (short)0, acc, false, false);
        }

        // Scatter D to LDS: D[M = r + mhalf][N = mrow]
#pragma unroll
        for (int r = 0; r < 8; ++r)
            sc[(r + mhalf) * 17 + mrow] = acc[r];
        __syncthreads();

        if (lane < 16) {
            // Row 'lane' of this tile: mask + serial top-4 (lowest-index ties)
            float s[NB];
#pragma unroll
            for (int n = 0; n < NB; ++n)
                s[n] = (n <= qb) ? sc[lane * 17 + n] : -__builtin_inff();
            unsigned chosen = 0;
            int idx[TOPK];
#pragma unroll
            for (int j = 0; j < TOPK; ++j) {
                float best = 0.f; int bi = -1;
#pragma unroll
                for (int n = 0; n < NB; ++n) {
                    if (!((chosen >> n) & 1u)) {
                        float v = s[n];
                        if (bi < 0 || v > best) { best = v; bi = n; }
                    }
                }
                chosen |= 1u << bi;
                idx[j] = bi;
            }
            int pos = rowBase + lane;
            ((int4*)topk_out)[(size_t)bh * S_ + pos] =
                make_int4(idx[0], idx[1], idx[2], idx[3]);
        }
        __syncthreads();
    }
}

// ---------------------------------------------------------------------------
// Kernel 3: query_block_indices[i] = i / 512
// ---------------------------------------------------------------------------
__global__ void qblock_kernel(int* __restrict__ out2) {
    int i = blockIdx.x * 256 + threadIdx.x;
    if (i < S_) out2[i] = i >> 9;
}

extern "C" void kernel_launch(void* const* d_in, const int* in_sizes, int n_in,
                              void* d_out, int out_size, void* d_ws, size_t ws_size,
                              hipStream_t stream) {
    const float* q = (const float*)d_in[0];
    const float* k = (const float*)d_in[1];
    // d_in[2] = seq_len scalar (fixed at 8192 for this problem shape)

    int*   out   = (int*)d_out;              // [B,H,S,4] int32 then [S] int32
    float* kmean = (float*)d_ws;             // B*H*NB*D floats = 512 KB

    kmean_kernel<<<B_ * H_ * NB, 128, 0, stream>>>(k, kmean);
    score_topk_kernel<<<B_ * H_ * NB, 128, 0, stream>>>(q, kmean, out);
    qblock_kernel<<<S_ / 256, 256, 0, stream>>>(out + (size_t)B_ * H_ * S_ * TOPK);
}